// HexDepthwiseConv_53772990546137
// MI455X (gfx1250) — compile-verified
//
#include <hip/hip_runtime.h>
#include <hip/hip_bf16.h>
#include <math.h>

typedef __attribute__((ext_vector_type(2))) float v2f;
typedef __attribute__((ext_vector_type(8))) float v8f;

#define B_ 32
#define N_ 4096
#define D_ 128
#define E_ 24576

// ---------------------------------------------------------------------------
// Kernel 0: zero the output (harness poisons d_out with 0xAA before timing).
// 64 MB of stores ~ 3 us at 23.3 TB/s.
// ---------------------------------------------------------------------------
__global__ void zero_out_kernel(float4* __restrict__ out, int n4) {
    int i = blockIdx.x * blockDim.x + threadIdx.x;
    if (i < n4) out[i] = make_float4(0.f, 0.f, 0.f, 0.f);
}

// ---------------------------------------------------------------------------
// Kernel 1: per-node scores via V_WMMA_F32_16X16X4_F32.
//   s_src[b,n] = x[b,n,:] . W[0:128]     (column N=0 of the B fragment)
//   s_dst[b,n] = x[b,n,:] . W[128:256]   (column N=1 of the B fragment)
// One wave per 16-row tile of the flattened (B*N) x D matrix; 32 accumulating
// WMMAs over K=128. Reads x exactly once (64 MB total).
//
// A-fragment layout (16x4 f32, ISA 7.12.2): lanes 0-15 -> rows M=0..15 with
// VGPR0=K0,VGPR1=K1; lanes 16-31 -> same rows with K2,K3.
// B-fragment (4x16): lane n carries column N=n; VGPR0/1 = K0/K1 (lanes 0-15)
// and K2/K3 (lanes 16-31).
// C/D (16x16 f32): VGPR r, lanes 0-15 -> (M=r, N=lane); lanes 16-31 ->
// (M=r+8, N=lane-16). So lanes 0/16 hold the s_src column, lanes 1/17 s_dst.
// ---------------------------------------------------------------------------
__global__ void node_score_kernel(const float* __restrict__ x,
                                  const float* __restrict__ W,
                                  float* __restrict__ ssrc,
                                  float* __restrict__ sdst) {
    const int lane   = threadIdx.x & 31;
    const int wave   = threadIdx.x >> 5;
    const int tile   = blockIdx.x * (blockDim.x >> 5) + wave; // over (B_*N_)/16
    const int rowBase = tile * 16;                            // flat b*N_+n row

    const int halfK = (lane >= 16) ? 2 : 0;                   // K sub-offset
    const int nsel  = lane & 15;                              // column N

    const float* __restrict__ xrow = x + (size_t)(rowBase + nsel) * D_ + halfK;
    // lanes with N==0 read Ws, N==1 read Wd; all other lanes read Ws but are
    // zeroed via bmask (branch-free: EXEC must stay all-1s around WMMA).
    const float* __restrict__ wcol = W + ((nsel == 1) ? D_ : 0) + halfK;
    const float bmask = (nsel <= 1) ? 1.0f : 0.0f;

    v8f acc = {};
#pragma unroll 4
    for (int k0 = 0; k0 < D_; k0 += 4) {
        v2f a, bf;
        a.x  = xrow[k0 + 0];
        a.y  = xrow[k0 + 1];
        bf.x = wcol[k0 + 0] * bmask;
        bf.y = wcol[k0 + 1] * bmask;
        acc = __builtin_amdgcn_wmma_f32_16x16x4_f32(
            /*neg_a=*/false, a, /*neg_b=*/false, bf,
            /*c_mod=*/(short)0, acc, /*reuse_a=*/false, /*reuse_b=*/false);
    }

    // Lanes 0/16: s_src for rows rowBase+{0..7}/{8..15}; lanes 1/17: s_dst.
    if (nsel <= 1) {
        float* __restrict__ dst = (nsel == 0) ? ssrc : sdst;
        const int mBase = rowBase + ((lane >= 16) ? 8 : 0);
#pragma unroll
        for (int r = 0; r < 8; ++r) dst[mBase + r] = acc[r];
    }
}

// ---------------------------------------------------------------------------
// Kernel 2: edge aggregation. One wave per (b,e); each lane owns a float4
// slice of the 128-wide row. x rows are gathered from L2 (x = 64 MB fits in
// the 192 MB L2), scatter via global_atomic_add_f32 (avg 6 edges per dst).
// ---------------------------------------------------------------------------
__global__ void edge_aggregate_kernel(const float* __restrict__ x,
                                      const int* __restrict__ edge_index,
                                      const float* __restrict__ bias,
                                      const float* __restrict__ ssrc,
                                      const float* __restrict__ sdst,
                                      float* __restrict__ out) {
    const int lane = threadIdx.x & 31;
    const int wave = (int)((blockIdx.x * (unsigned)blockDim.x + threadIdx.x) >> 5);
    if (wave >= B_ * E_) return;
    const int b = wave / E_;
    const int e = wave - b * E_;

    const int s = edge_index[e];        // src node
    const int d = edge_index[E_ + e];   // dst node

    const float sc  = ssrc[b * N_ + s] + sdst[b * N_ + d] + bias[0];
    const float att = 1.0f / (1.0f + __expf(-sc));

    const float4 xs = ((const float4*)(x + ((size_t)b * N_ + s) * D_))[lane];
    float* __restrict__ op = out + ((size_t)b * N_ + d) * D_ + lane * 4;
    atomicAdd(op + 0, xs.x * att);
    atomicAdd(op + 1, xs.y * att);
    atomicAdd(op + 2, xs.z * att);
    atomicAdd(op + 3, xs.w * att);
}

// ---------------------------------------------------------------------------
extern "C" void kernel_launch(void* const* d_in, const int* in_sizes, int n_in,
                              void* d_out, int out_size, void* d_ws, size_t ws_size,
                              hipStream_t stream) {
    const float* x          = (const float*)d_in[0];  // (B,N,D) f32
    const int*   edge_index = (const int*)  d_in[1];  // (2,E) i32
    const float* W          = (const float*)d_in[2];  // (1,2D) f32
    const float* bias       = (const float*)d_in[3];  // (1,) f32
    float* out = (float*)d_out;                       // (B,N,D) f32

    float* ssrc = (float*)d_ws;            // B_*N_ floats
    float* sdst = ssrc + (size_t)B_ * N_;  // B_*N_ floats

    // 1) zero out
    {
        const int n4 = (B_ * N_ * D_) / 4;
        zero_out_kernel<<<(n4 + 255) / 256, 256, 0, stream>>>((float4*)out, n4);
    }
    // 2) node scores via WMMA (8192 waves, 8 waves/block)
    {
        const int tiles  = (B_ * N_) / 16;
        const int wavesPerBlock = 8;
        node_score_kernel<<<tiles / wavesPerBlock, wavesPerBlock * 32, 0, stream>>>(
            x, W, ssrc, sdst);
    }
    // 3) edge aggregation (one wave per (b,e))
    {
        const long long totalThreads = (long long)B_ * E_ * 32;
        const int block = 256;
        const int grid  = (int)((totalThreads + block - 1) / block);
        edge_aggregate_kernel<<<grid, block, 0, stream>>>(
            x, edge_index, bias, ssrc, sdst, out);
    }
}